// MultiGrid_36455682409092
// MI455X (gfx1250) — compile-verified
//
#include <hip/hip_runtime.h>
#include <cstdint>
#include <cstddef>

// Native 4-wide float vector (supports arithmetic, .xyzw, and the
// nontemporal builtins; HIP's float4 class type does not).
typedef float v4f __attribute__((ext_vector_type(4)));

// ---------------------------------------------------------------------------
// Smoothstep weight
// ---------------------------------------------------------------------------
__device__ __forceinline__ float sstep(float t) { return t * t * (3.0f - 2.0f * t); }

// ---------------------------------------------------------------------------
// Trilinear corner indices + smoothstep weights for an S^3 volume,
// align_corners=true, border clamp (matches the JAX reference).
// ---------------------------------------------------------------------------
struct TriCoord {
  int i000, i001, i010, i011, i100, i101, i110, i111;
  float tx, ty, tz;
};

template <int S>
__device__ __forceinline__ TriCoord tricoord(float gx, float gy, float gz) {
  constexpr float SM1 = (float)(S - 1);
  constexpr float HS  = 0.5f * (float)(S - 1);
  float x = fminf(fmaxf((gx + 1.0f) * HS, 0.0f), SM1);
  float y = fminf(fmaxf((gy + 1.0f) * HS, 0.0f), SM1);
  float z = fminf(fmaxf((gz + 1.0f) * HS, 0.0f), SM1);
  float xf = floorf(x), yf = floorf(y), zf = floorf(z);
  int x0 = (int)xf, y0 = (int)yf, z0 = (int)zf;
  int x1 = x0 + 1; if (x1 > S - 1) x1 = S - 1;
  int y1 = y0 + 1; if (y1 > S - 1) y1 = S - 1;
  int z1 = z0 + 1; if (z1 > S - 1) z1 = S - 1;
  TriCoord c;
  c.tx = sstep(x - xf);
  c.ty = sstep(y - yf);
  c.tz = sstep(z - zf);
  int r00 = (z0 * S + y0) * S;
  int r01 = (z0 * S + y1) * S;
  int r10 = (z1 * S + y0) * S;
  int r11 = (z1 * S + y1) * S;
  c.i000 = r00 + x0; c.i001 = r00 + x1;
  c.i010 = r01 + x0; c.i011 = r01 + x1;
  c.i100 = r10 + x0; c.i101 = r10 + x1;
  c.i110 = r11 + x0; c.i111 = r11 + x1;
  return c;
}

__device__ __forceinline__ v4f trilerp(v4f c000, v4f c001, v4f c010, v4f c011,
                                       v4f c100, v4f c101, v4f c110, v4f c111,
                                       float tx, float ty, float tz) {
  v4f c00 = c000 + (c001 - c000) * tx;
  v4f c01 = c010 + (c011 - c010) * tx;
  v4f c10 = c100 + (c101 - c100) * tx;
  v4f c11 = c110 + (c111 - c110) * tx;
  v4f c0  = c00 + (c01 - c00) * ty;
  v4f c1  = c10 + (c11 - c10) * ty;
  return c0 + (c1 - c0) * tz;
}

// ---------------------------------------------------------------------------
// Fast path: channel-interleaved [D,H,W,4] volume -> one b128 load per corner.
// NT = true marks loads non-temporal (for the 256^3 level that exceeds the
// hot L2 working set and has ~zero reuse).
// ---------------------------------------------------------------------------
template <int S, bool NT>
__device__ __forceinline__ v4f vload(const v4f* __restrict__ p, int idx) {
  if constexpr (NT) {
    return __builtin_nontemporal_load(p + idx);
  } else {
    return p[idx];
  }
}

template <int S, bool NT>
__device__ __forceinline__ v4f sample_vec(const v4f* __restrict__ vol,
                                          float gx, float gy, float gz) {
  TriCoord c = tricoord<S>(gx, gy, gz);
  v4f c000 = vload<S, NT>(vol, c.i000);
  v4f c001 = vload<S, NT>(vol, c.i001);
  v4f c010 = vload<S, NT>(vol, c.i010);
  v4f c011 = vload<S, NT>(vol, c.i011);
  v4f c100 = vload<S, NT>(vol, c.i100);
  v4f c101 = vload<S, NT>(vol, c.i101);
  v4f c110 = vload<S, NT>(vol, c.i110);
  v4f c111 = vload<S, NT>(vol, c.i111);
  return trilerp(c000, c001, c010, c011, c100, c101, c110, c111, c.tx, c.ty, c.tz);
}

// ---------------------------------------------------------------------------
// Fallback path: channel-major [4,D,H,W] (original layout), 4 scalar loads
// per corner.
// ---------------------------------------------------------------------------
template <int S>
__device__ __forceinline__ v4f corner_ch(const float* __restrict__ vol, int idx) {
  constexpr int DHW = S * S * S;
  v4f r;
  r.x = vol[idx];
  r.y = vol[DHW + idx];
  r.z = vol[2 * DHW + idx];
  r.w = vol[3 * DHW + idx];
  return r;
}

template <int S>
__device__ __forceinline__ v4f sample_ch(const float* __restrict__ vol,
                                         float gx, float gy, float gz) {
  TriCoord c = tricoord<S>(gx, gy, gz);
  v4f c000 = corner_ch<S>(vol, c.i000);
  v4f c001 = corner_ch<S>(vol, c.i001);
  v4f c010 = corner_ch<S>(vol, c.i010);
  v4f c011 = corner_ch<S>(vol, c.i011);
  v4f c100 = corner_ch<S>(vol, c.i100);
  v4f c101 = corner_ch<S>(vol, c.i101);
  v4f c110 = corner_ch<S>(vol, c.i110);
  v4f c111 = corner_ch<S>(vol, c.i111);
  return trilerp(c000, c001, c010, c011, c100, c101, c110, c111, c.tx, c.ty, c.tz);
}

// ---------------------------------------------------------------------------
// Stage this block's 256 xyz triplets (768 floats, fully contiguous) into LDS.
// gfx1250 path: GLOBAL_LOAD_ASYNC_TO_LDS_B32 + s_wait_asynccnt (ASYNCcnt).
// Builtin prototype on this toolchain: (int AS1*, int AS3*, imm offset, imm cpol).
// ---------------------------------------------------------------------------
__device__ __forceinline__ void stage_grid(const float* __restrict__ grid,
                                           float* sh, int base, int n) {
  int todo = 3 * (n - base);
  if (todo > 3 * 256) todo = 3 * 256;
  const float* gsrc = grid + (size_t)base * 3;
#if defined(__gfx1250__) && __has_builtin(__builtin_amdgcn_global_load_async_to_lds_b32)
  for (int i = threadIdx.x; i < todo; i += 256) {
    __builtin_amdgcn_global_load_async_to_lds_b32(
        (__attribute__((address_space(1))) int*)(gsrc + i),
        (__attribute__((address_space(3))) int*)(sh + i),
        0, 0);
  }
#if __has_builtin(__builtin_amdgcn_s_wait_asynccnt)
  __builtin_amdgcn_s_wait_asynccnt(0);
#else
  asm volatile("s_wait_asynccnt 0" ::: "memory");
#endif
#else
  for (int i = threadIdx.x; i < todo; i += 256) sh[i] = gsrc[i];
#endif
  __syncthreads();
}

// ---------------------------------------------------------------------------
// Layout transform: [4,S,S,S] channel-major -> [S,S,S,4] interleaved.
// Reads 4 coalesced streams, writes coalesced b128.
// ---------------------------------------------------------------------------
template <int S>
__global__ void __launch_bounds__(256) interleave_k(const float* __restrict__ src,
                                                    v4f* __restrict__ dst) {
  constexpr int DHW = S * S * S;
  int v = blockIdx.x * 256 + threadIdx.x;
  if (v >= DHW) return;
  v4f o;
  o.x = src[v];
  o.y = src[DHW + v];
  o.z = src[2 * DHW + v];
  o.w = src[3 * DHW + v];
  dst[v] = o;
}

// ---------------------------------------------------------------------------
// Main sampler (fast, interleaved volumes). Output layout: out[(lvl*4+c)*n + p].
// ---------------------------------------------------------------------------
__global__ void __launch_bounds__(256) sample_vec_k(
    const float* __restrict__ grid,
    const v4f* __restrict__ t0, const v4f* __restrict__ t1,
    const v4f* __restrict__ t2, const v4f* __restrict__ t3,
    float* __restrict__ out, int n) {
  __shared__ float sh[3 * 256];
  int base = blockIdx.x * 256;
  stage_grid(grid, sh, base, n);
  int p = base + threadIdx.x;
  if (p >= n) return;
  float gx = sh[3 * threadIdx.x + 0];
  float gy = sh[3 * threadIdx.x + 1];
  float gz = sh[3 * threadIdx.x + 2];

  v4f f0 = sample_vec<32,  false>(t0, gx, gy, gz);
  v4f f1 = sample_vec<64,  false>(t1, gx, gy, gz);
  v4f f2 = sample_vec<128, false>(t2, gx, gy, gz);
  v4f f3 = sample_vec<256, true >(t3, gx, gy, gz);  // NT: 256MB stream, no reuse

  size_t np = (size_t)n;
  float* o = out + p;
  __builtin_nontemporal_store(f0.x, o + 0 * np);
  __builtin_nontemporal_store(f0.y, o + 1 * np);
  __builtin_nontemporal_store(f0.z, o + 2 * np);
  __builtin_nontemporal_store(f0.w, o + 3 * np);
  __builtin_nontemporal_store(f1.x, o + 4 * np);
  __builtin_nontemporal_store(f1.y, o + 5 * np);
  __builtin_nontemporal_store(f1.z, o + 6 * np);
  __builtin_nontemporal_store(f1.w, o + 7 * np);
  __builtin_nontemporal_store(f2.x, o + 8 * np);
  __builtin_nontemporal_store(f2.y, o + 9 * np);
  __builtin_nontemporal_store(f2.z, o + 10 * np);
  __builtin_nontemporal_store(f2.w, o + 11 * np);
  __builtin_nontemporal_store(f3.x, o + 12 * np);
  __builtin_nontemporal_store(f3.y, o + 13 * np);
  __builtin_nontemporal_store(f3.z, o + 14 * np);
  __builtin_nontemporal_store(f3.w, o + 15 * np);
}

// ---------------------------------------------------------------------------
// Fallback sampler (original channel-major volumes, no workspace needed).
// ---------------------------------------------------------------------------
__global__ void __launch_bounds__(256) sample_ch_k(
    const float* __restrict__ grid,
    const float* __restrict__ v0, const float* __restrict__ v1,
    const float* __restrict__ v2, const float* __restrict__ v3,
    float* __restrict__ out, int n) {
  __shared__ float sh[3 * 256];
  int base = blockIdx.x * 256;
  stage_grid(grid, sh, base, n);
  int p = base + threadIdx.x;
  if (p >= n) return;
  float gx = sh[3 * threadIdx.x + 0];
  float gy = sh[3 * threadIdx.x + 1];
  float gz = sh[3 * threadIdx.x + 2];

  v4f f0 = sample_ch<32 >(v0, gx, gy, gz);
  v4f f1 = sample_ch<64 >(v1, gx, gy, gz);
  v4f f2 = sample_ch<128>(v2, gx, gy, gz);
  v4f f3 = sample_ch<256>(v3, gx, gy, gz);

  size_t np = (size_t)n;
  float* o = out + p;
  o[0 * np] = f0.x;  o[1 * np] = f0.y;  o[2 * np]  = f0.z;  o[3 * np]  = f0.w;
  o[4 * np] = f1.x;  o[5 * np] = f1.y;  o[6 * np]  = f1.z;  o[7 * np]  = f1.w;
  o[8 * np] = f2.x;  o[9 * np] = f2.y;  o[10 * np] = f2.z;  o[11 * np] = f2.w;
  o[12 * np] = f3.x; o[13 * np] = f3.y; o[14 * np] = f3.z;  o[15 * np] = f3.w;
}

// ---------------------------------------------------------------------------
// Host launcher
// ---------------------------------------------------------------------------
extern "C" void kernel_launch(void* const* d_in, const int* in_sizes, int n_in,
                              void* d_out, int out_size, void* d_ws, size_t ws_size,
                              hipStream_t stream) {
  const float* grid = (const float*)d_in[0];
  const float* v0   = (const float*)d_in[1];  // [1,4,32,32,32]
  const float* v1   = (const float*)d_in[2];  // [1,4,64,64,64]
  const float* v2   = (const float*)d_in[3];  // [1,4,128,128,128]
  const float* v3   = (const float*)d_in[4];  // [1,4,256,256,256]
  float* out = (float*)d_out;

  int n = in_sizes[0] / 3;           // 1,048,576 points
  int blocks = (n + 255) / 256;

  // Interleaved workspace layout (v4f elements):
  //   t0 @ 0        (32^3  = 32768)
  //   t1 @ 32768    (64^3  = 262144)
  //   t2 @ 294912   (128^3 = 2097152)
  //   t3 @ 2392064  (256^3 = 16777216)
  const size_t need_bytes = 19169280ull * sizeof(v4f);  // 306,708,480 B

  if (d_ws != nullptr && ws_size >= need_bytes) {
    v4f* ws = (v4f*)d_ws;
    v4f* t0 = ws;
    v4f* t1 = ws + 32768;
    v4f* t2 = ws + 294912;
    v4f* t3 = ws + 2392064;
    interleave_k<32 ><<<(32 * 32 * 32) / 256, 256, 0, stream>>>(v0, t0);
    interleave_k<64 ><<<(64 * 64 * 64) / 256, 256, 0, stream>>>(v1, t1);
    interleave_k<128><<<(128 * 128 * 128) / 256, 256, 0, stream>>>(v2, t2);
    interleave_k<256><<<(256 * 256 * 256) / 256, 256, 0, stream>>>(v3, t3);
    sample_vec_k<<<blocks, 256, 0, stream>>>(grid, t0, t1, t2, t3, out, n);
  } else {
    sample_ch_k<<<blocks, 256, 0, stream>>>(grid, v0, v1, v2, v3, out, n);
  }
}